// DiscreteStateSpaceModel_77077483094247
// MI455X (gfx1250) — compile-verified
//
#include <hip/hip_runtime.h>

// Problem constants (from reference)
#define B_SZ   16
#define SEQ    1024
#define D_IN_  512
#define H_     256
#define D_OUT_ 512

// Persistent-kernel shape: 256 waves = one wave per 16x16 tile of a 256x256 GEMM
#define NBLK   32
#define NTHR   256

typedef __attribute__((ext_vector_type(2))) float v2f;
typedef __attribute__((ext_vector_type(8))) float v8f;

__device__ __forceinline__ v8f wmma4(v2f a, v2f b, v8f c) {
    return __builtin_amdgcn_wmma_f32_16x16x4_f32(false, a, false, b, (short)0, c, false, false);
}

// ---------------------------------------------------------------------------
// One 16x16 output tile of Cout = A @ op(B) (+bias), V_WMMA_F32_16X16X4_F32,
// K-reduction split over 4 independent accumulator chains (K % 16 == 0).
// Fragment layouts per CDNA5 ISA 7.12.2 (wave32):
//   A 16x4 : lane r=lane&15, hi=lane>>4 : {A[r][k+2hi], A[r][k+2hi+1]}
//   B 4x16 : lane c=lane&15, hi        : {B[k+2hi][c], B[k+2hi+1][c]}
//   C 16x16: VGPR i, lanes 0-15 -> row i, lanes 16-31 -> row i+8
// ---------------------------------------------------------------------------
template<int BT, int BIAS>
__device__ __forceinline__
void tile_gemm(int tm, int tn, int lane,
               const float* __restrict__ A, int lda,
               const float* __restrict__ B, int ldb,
               const float* __restrict__ bias,
               float* __restrict__ Cst, int ldc, int K)
{
    const int r  = lane & 15;
    const int hi = lane >> 4;

    const float* Arow = A + (long)(tm * 16 + r) * lda + 2 * hi;

    const v8f vzero = (v8f){0.f, 0.f, 0.f, 0.f, 0.f, 0.f, 0.f, 0.f};
    v8f acc0 = vzero, acc1 = vzero, acc2 = vzero, acc3 = vzero;

    if (BT) {
        // B stored [N,K]: contiguous K pairs (b64 loads)
        const float* Brow = B + (long)(tn * 16 + r) * ldb + 2 * hi;
        for (int k = 0; k < K; k += 16) {
            v2f a0, a1, a2, a3, b0, b1, b2, b3;
            a0.x = Arow[k];      a0.y = Arow[k + 1];
            a1.x = Arow[k + 4];  a1.y = Arow[k + 5];
            a2.x = Arow[k + 8];  a2.y = Arow[k + 9];
            a3.x = Arow[k + 12]; a3.y = Arow[k + 13];
            b0.x = Brow[k];      b0.y = Brow[k + 1];
            b1.x = Brow[k + 4];  b1.y = Brow[k + 5];
            b2.x = Brow[k + 8];  b2.y = Brow[k + 9];
            b3.x = Brow[k + 12]; b3.y = Brow[k + 13];
            acc0 = wmma4(a0, b0, acc0);
            acc1 = wmma4(a1, b1, acc1);
            acc2 = wmma4(a2, b2, acc2);
            acc3 = wmma4(a3, b3, acc3);
        }
    } else {
        // B stored [K,N]
        const float* Bcol = B + (long)(2 * hi) * ldb + tn * 16 + r;
        for (int k = 0; k < K; k += 16) {
            v2f a0, a1, a2, a3, b0, b1, b2, b3;
            a0.x = Arow[k];      a0.y = Arow[k + 1];
            a1.x = Arow[k + 4];  a1.y = Arow[k + 5];
            a2.x = Arow[k + 8];  a2.y = Arow[k + 9];
            a3.x = Arow[k + 12]; a3.y = Arow[k + 13];
            const float* bp0 = Bcol + (long)(k)      * ldb;
            const float* bp1 = Bcol + (long)(k + 4)  * ldb;
            const float* bp2 = Bcol + (long)(k + 8)  * ldb;
            const float* bp3 = Bcol + (long)(k + 12) * ldb;
            b0.x = bp0[0]; b0.y = bp0[ldb];
            b1.x = bp1[0]; b1.y = bp1[ldb];
            b2.x = bp2[0]; b2.y = bp2[ldb];
            b3.x = bp3[0]; b3.y = bp3[ldb];
            acc0 = wmma4(a0, b0, acc0);
            acc1 = wmma4(a1, b1, acc1);
            acc2 = wmma4(a2, b2, acc2);
            acc3 = wmma4(a3, b3, acc3);
        }
    }

    v8f acc = (acc0 + acc1) + (acc2 + acc3);

    float bs = 0.f;
    if (BIAS) bs = bias[tn * 16 + r];

    #pragma unroll
    for (int i = 0; i < 8; ++i) {
        const int row = tm * 16 + i + 8 * hi;
        Cst[(long)row * ldc + tn * 16 + r] = acc[i] + bs;
    }
}

// ---------------------------------------------------------------------------
// Device-wide generation barrier (agent scope). All NBLK blocks co-resident
// (32 blocks x 256 thr on a >=100-WGP part), so spinning is safe.
// count/gen are zeroed by hipMemsetAsync before each launch.
// ---------------------------------------------------------------------------
__device__ __forceinline__ void grid_sync(unsigned* cnt, unsigned* gen)
{
    __syncthreads();
    if (threadIdx.x == 0) {
        __threadfence();  // release prior global writes to device scope
        unsigned g = __hip_atomic_load(gen, __ATOMIC_RELAXED, __HIP_MEMORY_SCOPE_AGENT);
        unsigned a = __hip_atomic_fetch_add(cnt, 1u, __ATOMIC_ACQ_REL, __HIP_MEMORY_SCOPE_AGENT);
        if (a == (unsigned)(NBLK - 1)) {
            __hip_atomic_store(cnt, 0u, __ATOMIC_RELAXED, __HIP_MEMORY_SCOPE_AGENT);
            __hip_atomic_store(gen, g + 1u, __ATOMIC_RELEASE, __HIP_MEMORY_SCOPE_AGENT);
        } else {
            while (__hip_atomic_load(gen, __ATOMIC_ACQUIRE, __HIP_MEMORY_SCOPE_AGENT) == g) {
                __builtin_amdgcn_s_sleep(8);
            }
        }
        __threadfence();  // acquire: invalidate stale near caches
    }
    __syncthreads();
}

// ---------------------------------------------------------------------------
// Single persistent kernel:
//   u_ws  = u_t[:,SEQ-1,:] @ W_in^T + b_in      (waves 0..15, pre-barrier)
//   P     = A_d^1023 (square-and-multiply, 9 rounds of R = R^2 * A_d)
//   Klast = C @ (P @ B_d)
//   out   = (u_ws @ Klast) @ W_out^T + b_out
// ---------------------------------------------------------------------------
__global__ __launch_bounds__(NTHR)
void ssm_fused_kernel(const float* __restrict__ u_t,
                      const float* __restrict__ W_in,
                      const float* __restrict__ b_in,
                      const float* __restrict__ Cm,
                      const float* __restrict__ W_out,
                      const float* __restrict__ b_out,
                      const float* __restrict__ A_d,
                      const float* __restrict__ B_d,
                      float* __restrict__ buf0,
                      float* __restrict__ buf1,
                      float* __restrict__ u_ws,
                      float* __restrict__ y_ws,
                      unsigned* __restrict__ bar,
                      float* __restrict__ out)
{
    const int wid  = (int)((blockIdx.x * NTHR + threadIdx.x) >> 5);  // 0..255
    const int lane = (int)(threadIdx.x & 31);
    unsigned* cnt = bar;
    unsigned* gen = bar + 1;

    // ---- Input projection, overlapped with the chain start (no barrier:
    //      u_ws is first consumed ~20 barriers from now). 16 tiles.
    if (wid < 16) {
        tile_gemm<1, 1>(0, wid, lane,
                        u_t + (size_t)(SEQ - 1) * D_IN_, SEQ * D_IN_,
                        W_in, D_IN_, b_in, u_ws, H_, D_IN_);
    }

    // ---- Power chain: R = A_d (e=1); 9 rounds of R = R^2 * A_d => e = 2^10-1
    const int tm = wid >> 4;     // 16x16 tile grid over 256x256
    const int tn = wid & 15;
    for (int i = 0; i < 9; ++i) {
        const float* S = (i == 0) ? A_d : buf0;
        tile_gemm<0, 0>(tm, tn, lane, S, H_, S, H_, nullptr, buf1, H_, H_);
        grid_sync(cnt, gen);
        tile_gemm<0, 0>(tm, tn, lane, buf1, H_, A_d, H_, nullptr, buf0, H_, H_);
        grid_sync(cnt, gen);
    }

    // ---- M = P @ B_d
    tile_gemm<0, 0>(tm, tn, lane, buf0, H_, B_d, H_, nullptr, buf1, H_, H_);
    grid_sync(cnt, gen);

    // ---- K_last = C @ M
    tile_gemm<0, 0>(tm, tn, lane, Cm, H_, buf1, H_, nullptr, buf0, H_, H_);
    grid_sync(cnt, gen);

    // ---- y = u_last @ K_last   [16 x 256] -> 16 tiles
    if (wid < 16) {
        tile_gemm<0, 0>(0, wid, lane, u_ws, H_, buf0, H_, nullptr, y_ws, H_, H_);
    }
    grid_sync(cnt, gen);

    // ---- out = y @ W_out^T + b_out   [16 x 512] -> 32 tiles
    if (wid < 32) {
        tile_gemm<1, 1>(0, wid, lane, y_ws, H_, W_out, H_, b_out, out, D_OUT_, H_);
    }
}

extern "C" void kernel_launch(void* const* d_in, const int* in_sizes, int n_in,
                              void* d_out, int out_size, void* d_ws, size_t ws_size,
                              hipStream_t stream)
{
    (void)in_sizes; (void)n_in; (void)out_size; (void)ws_size;

    const float* u_t   = (const float*)d_in[0];  // [B, SEQ, D_IN]
    const float* W_in  = (const float*)d_in[1];  // [H, D_IN]
    const float* b_in  = (const float*)d_in[2];  // [H]
    const float* Cm    = (const float*)d_in[3];  // [H, H]
    const float* W_out = (const float*)d_in[4];  // [D_OUT, H]
    const float* b_out = (const float*)d_in[5];  // [D_OUT]
    const float* A_d   = (const float*)d_in[6];  // [H, H]
    const float* B_d   = (const float*)d_in[7];  // [H, H]
    float* out = (float*)d_out;                  // [B, D_OUT]

    const size_t matElems = (size_t)H_ * H_;     // 65536 floats

    // Workspace carving (~545 KB total)
    float*    buf0 = (float*)d_ws;                       // 256 KB
    float*    buf1 = buf0 + matElems;                    // 256 KB
    float*    u_ws = buf1 + matElems;                    // 16 KB  [B, H]
    float*    y_ws = u_ws + (size_t)B_SZ * H_;           // 16 KB  [B, H]
    unsigned* bar  = (unsigned*)(y_ws + (size_t)B_SZ * H_);  // 2 x u32

    // Barrier state must start at 0 every call (ws is poisoned once by the
    // harness and our kernel leaves gen incremented).
    hipMemsetAsync(bar, 0, 2 * sizeof(unsigned), stream);

    hipLaunchKernelGGL(ssm_fused_kernel, dim3(NBLK), dim3(NTHR), 0, stream,
                       u_t, W_in, b_in, Cm, W_out, b_out, A_d, B_d,
                       buf0, buf1, u_ws, y_ws, bar, out);
}